// AccumulativeTransform_59545426592335
// MI455X (gfx1250) — compile-verified
//
#include <hip/hip_runtime.h>

typedef __attribute__((ext_vector_type(2))) float v2f;
typedef __attribute__((ext_vector_type(4))) float v4f;
typedef __attribute__((ext_vector_type(8))) float v8f;

#define T_FRAMES   1024
#define DCH        4
#define ROW_FLOATS (T_FRAMES * DCH)   // 4096 floats = 16 KB per (b,c) row
#define SEG        64                 // 64 segments of 16 frames
#define SEG_STRIDE 76                 // padded floats per segment in LDS (bank-conflict free)
#define BLOCK      128                // 4 waves

// LDS byte offset of a __shared__ object (generic -> addrspace(3) -> 32-bit offset)
__device__ __forceinline__ unsigned lds_off(const void* p) {
    return (unsigned)(unsigned long long)(const __attribute__((address_space(3))) void*)p;
}

// Inclusive prefix sum along frames via triangular-matrix WMMA:
//   D(16x16) = L(16x16, lower-tri ones) * B(16x16)
// realized as 4 chained V_WMMA_F32_16X16X4_F32 (K-panels of L).
// Columns of B = 4 segments x 4 d-channels; one wave handles 4 such groups.
// Staging uses CDNA5 async Global->LDS DMA (ASYNCcnt), output uses NT stores.
__global__ __launch_bounds__(BLOCK)
void AccumulativeTransform_59545426592335_kernel(const float* __restrict__ x,
                                                 float* __restrict__ y) {
    __shared__ __align__(16) float buf[SEG * SEG_STRIDE];   // padded row staging / scan buffer
    __shared__ __align__(16) float totals[SEG * DCH];       // per-segment totals
    __shared__ __align__(16) float offs[SEG * DCH];         // exclusive cross-segment offsets

    const int tid  = threadIdx.x;
    const int lane = tid & 31;
    const int wave = tid >> 5;

    const long long rowBase = (long long)blockIdx.x * ROW_FLOATS;
    const float* __restrict__ xrow = x + rowBase;
    v4f*         __restrict__ yout = (v4f*)(y + rowBase);

    // ---- Phase 0: async DMA of the 16KB row into padded LDS layout ----
    // GVS form: gaddr = saddr(row base) + vaddr(byte offset); vdst = LDS byte addr.
#pragma unroll
    for (int k = 0; k < 8; ++k) {
        int t = tid + k * BLOCK;          // frame index 0..1023
        int s = t >> 4, tl = t & 15;
        unsigned ldsb = lds_off(&buf[s * SEG_STRIDE + tl * 4]);
        unsigned goff = (unsigned)t * 16u;
        asm volatile("global_load_async_to_lds_b128 %0, %1, %2"
                     :: "v"(ldsb), "v"(goff), "s"(xrow) : "memory");
    }
    asm volatile("s_wait_asynccnt 0x0" ::: "memory");   // our wave's DMA landed in LDS
    __syncthreads();                                     // ...visible to all waves

    // ---- Phase 1: intra-segment inclusive scans on the matrix unit ----
    // A-matrix (16x4 f32) lane layout: M = lane&15; VGPR0 holds K={0|2}, VGPR1 K={1|3}
    // selected by hi = lane>>4.  B mirrored: N = lane&15, rows K likewise.
    const int hi = lane >> 4;
    const int n  = lane & 15;

    v2f A[4];
#pragma unroll
    for (int j = 0; j < 4; ++j) {         // K-panel j covers K = 4j..4j+3 of L
        int k0 = 4 * j + 2 * hi;
        A[j].x = (k0     <= n) ? 1.0f : 0.0f;   // L[m][k] = (k <= m)
        A[j].y = (k0 + 1 <= n) ? 1.0f : 0.0f;
    }
    // This lane's column base inside a 4-segment group: seg-sub = n>>2, d = n&3
    const int colBase = (n >> 2) * SEG_STRIDE + (n & 3);

#pragma unroll
    for (int gi = 0; gi < 4; ++gi) {
        int g  = wave * 4 + gi;                       // group of 4 segments (0..15)
        int gb = 4 * g * SEG_STRIDE + colBase;        // lane's column base address
        v8f acc = {};
#pragma unroll
        for (int j = 0; j < 4; ++j) {
            int tl0 = 4 * j + 2 * hi;                 // frame-within-segment (K dim)
            v2f b;
            b.x = buf[gb + tl0 * 4];
            b.y = buf[gb + tl0 * 4 + 4];
            acc = __builtin_amdgcn_wmma_f32_16x16x4_f32(
                      false, A[j], false, b, (short)0, acc, false, false);
        }
        // D lane layout: element (M = r + 8*hi, N = n) in acc[r]; write back in place.
#pragma unroll
        for (int r = 0; r < 8; ++r) {
            int m = r + 8 * hi;
            buf[gb + m * 4] = acc[r];
        }
        if (hi) {                                     // M=15 row = segment totals
            int s = 4 * g + (n >> 2);
            totals[s * 4 + (n & 3)] = acc[7];
        }
    }
    __syncthreads();

    // ---- Phase 2: exclusive scan of segment totals, one chain per d-channel ----
    if (tid < DCH) {
        float run = 0.0f;
        for (int s = 0; s < SEG; ++s) {
            offs[s * 4 + tid] = run;
            run += totals[s * 4 + tid];
        }
    }
    __syncthreads();

    // ---- Phase 3: add cross-segment offsets, coalesced non-temporal b128 stores ----
#pragma unroll
    for (int k = 0; k < 8; ++k) {
        int t = tid + k * BLOCK;
        int s = t >> 4, tl = t & 15;
        v4f v = *(const v4f*)(&buf[s * SEG_STRIDE + tl * 4]);
        v4f o = *(const v4f*)(&offs[s * 4]);
        __builtin_nontemporal_store(v + o, &yout[t]);
    }
}

extern "C" void kernel_launch(void* const* d_in, const int* in_sizes, int n_in,
                              void* d_out, int out_size, void* d_ws, size_t ws_size,
                              hipStream_t stream) {
    (void)n_in; (void)d_ws; (void)ws_size; (void)out_size;
    const float* x = (const float*)d_in[0];
    float* y = (float*)d_out;
    int rows = in_sizes[0] / ROW_FLOATS;   // 64*256 = 16384 rows of 1024x4
    AccumulativeTransform_59545426592335_kernel<<<dim3(rows), dim3(BLOCK), 0, stream>>>(x, y);
}